// LIFSpike_35107062678294
// MI455X (gfx1250) — compile-verified
//
#include <hip/hip_runtime.h>
#include <cstdint>

// LIF spike scan: x[B=32,T=16,C=128,H=32,W=32] f32 -> spikes same shape.
// Pure streaming (512 MB traffic, ~22us at 23.3 TB/s). Implemented as a
// double-buffered async-copy pipeline: global_load_async_to_lds_b128 stages
// tiles into LDS (ASYNCcnt tracked), while the wave computes the previous
// tile and streams spikes out with nontemporal 128-bit stores.

typedef __attribute__((ext_vector_type(4))) float v4f;

namespace {
constexpr int   TT        = 16;                 // time steps (scan length)
constexpr int   BLOCK     = 128;                // threads per workgroup
constexpr int   CHW       = 128 * 32 * 32;      // 131072 elems per (b,t) plane
constexpr int   BATCH     = 32;
constexpr int   CHUNKS    = CHW / (4 * BLOCK);  // 256 float4-chunks per plane
constexpr int   NUM_TILES = BATCH * CHUNKS;     // 8192 tiles
constexpr int   GRID      = 2048;               // persistent workgroups
constexpr float TAU       = 0.25f;
constexpr float V_TH      = 1.0f;
}

// Issue 16 async b128 loads (one per timestep) for this thread's float4 chain
// into the given LDS buffer. Read-once input -> NT temporal hint.
__device__ __forceinline__ void lif_async_tile_load(const float* __restrict__ x,
                                                    int tile, int tid,
                                                    v4f (*buf)[BLOCK]) {
  const int b     = tile / CHUNKS;
  const int chunk = tile - b * CHUNKS;
  const size_t base = (size_t)b * ((size_t)TT * CHW) +
                      (size_t)chunk * (BLOCK * 4) + (size_t)tid * 4;
#pragma unroll
  for (int t = 0; t < TT; ++t) {
    uint64_t gaddr = (uint64_t)(uintptr_t)(x + base + (size_t)t * CHW);
    uint32_t laddr = (uint32_t)(uintptr_t)(&buf[t][tid]);
    asm volatile("global_load_async_to_lds_b128 %0, %1, off th:TH_LOAD_NT"
                 :: "v"(laddr), "v"(gaddr)
                 : "memory");
  }
}

__global__ void __launch_bounds__(BLOCK)
lif_scan_kernel(const float* __restrict__ x, float* __restrict__ out) {
  // Double buffer: 2 x (16 t-planes x 128 lanes x 16B) = 64 KB LDS.
  // Each thread reads back only the slots it wrote -> no barriers needed.
  __shared__ v4f sbuf[2][TT][BLOCK];

  const int tid = threadIdx.x;
  int cur  = 0;
  int tile = (int)blockIdx.x;

  if (tile < NUM_TILES) lif_async_tile_load(x, tile, tid, sbuf[0]);

  for (; tile < NUM_TILES; tile += GRID) {
    const int  nxt      = tile + GRID;
    const bool prefetch = (nxt < NUM_TILES);

    if (prefetch) {
      // Kick off next tile's 16 loads into the other buffer, then wait for
      // the *current* tile only: async loads complete in order, so
      // ASYNCcnt <= 16 means the first 16 (current tile) have landed.
      lif_async_tile_load(x, nxt, tid, sbuf[cur ^ 1]);
      asm volatile("s_wait_asynccnt 0x10" ::: "memory");
    } else {
      asm volatile("s_wait_asynccnt 0x0" ::: "memory");
    }

    const int b     = tile / CHUNKS;
    const int chunk = tile - b * CHUNKS;
    const size_t base = (size_t)b * ((size_t)TT * CHW) +
                        (size_t)chunk * (BLOCK * 4) + (size_t)tid * 4;

    v4f mem = (v4f)(0.0f);
#pragma unroll
    for (int t = 0; t < TT; ++t) {
      v4f xt = sbuf[cur][t][tid];          // ds_load_b128

      mem.x = fmaf(TAU, mem.x, xt.x);
      mem.y = fmaf(TAU, mem.y, xt.y);
      mem.z = fmaf(TAU, mem.z, xt.z);
      mem.w = fmaf(TAU, mem.w, xt.w);

      v4f s;
      s.x = (mem.x > V_TH) ? 1.0f : 0.0f;
      s.y = (mem.y > V_TH) ? 1.0f : 0.0f;
      s.z = (mem.z > V_TH) ? 1.0f : 0.0f;
      s.w = (mem.w > V_TH) ? 1.0f : 0.0f;

      // Streaming output, never re-read -> nontemporal b128 store.
      __builtin_nontemporal_store(
          s, reinterpret_cast<v4f*>(out + base + (size_t)t * CHW));

      // Hard reset: mem *= (1 - spike)
      mem.x = (mem.x > V_TH) ? 0.0f : mem.x;
      mem.y = (mem.y > V_TH) ? 0.0f : mem.y;
      mem.z = (mem.z > V_TH) ? 0.0f : mem.z;
      mem.w = (mem.w > V_TH) ? 0.0f : mem.w;
    }
    cur ^= 1;
  }
}

extern "C" void kernel_launch(void* const* d_in, const int* in_sizes, int n_in,
                              void* d_out, int out_size, void* d_ws, size_t ws_size,
                              hipStream_t stream) {
  (void)in_sizes; (void)n_in; (void)d_ws; (void)ws_size; (void)out_size;
  const float* x   = (const float*)d_in[0];
  float*       out = (float*)d_out;
  lif_scan_kernel<<<GRID, BLOCK, 0, stream>>>(x, out);
}